// DistMultDecoder_65601330479214
// MI455X (gfx1250) — compile-verified
//
#include <hip/hip_runtime.h>
#include <hip/hip_bf16.h>
#include <math.h>

typedef float v2f __attribute__((ext_vector_type(2)));
typedef float v8f __attribute__((ext_vector_type(8)));

#define HID 128

// ---------------------------------------------------------------------------
// Phase 1: y = z @ W^T   (y[m][n] = sum_k z[m][k] * W[n][k])
// One block = 256 threads = 8 waves; block owns 16 rows of z (staged in LDS),
// wave w owns output columns [16w, 16w+16). K=128 stepped by 4 via
// V_WMMA_F32_16X16X4_F32 (fp32 in / fp32 accumulate -> bit-compatible math).
// ---------------------------------------------------------------------------
__global__ __launch_bounds__(256)
void distmult_gemm_zWT(const float* __restrict__ z,
                       const float* __restrict__ W,
                       float* __restrict__ y, int nrows)
{
    __shared__ float As[16 * HID];            // 16 rows x 128 cols = 8 KB
    const int m0  = blockIdx.x * 16;
    const int tid = threadIdx.x;

    // Cooperative coalesced load of the A tile (512 float4 chunks, 2/thread).
    #pragma unroll
    for (int c = tid; c < 512; c += 256) {
        const int row  = c >> 5;              // 32 float4 per row
        const int col4 = c & 31;
        int grow = m0 + row;
        if (grow >= nrows) grow = nrows - 1;  // clamp tail (N%16==0 in practice)
        const float4 v = *(const float4*)(z + (size_t)grow * HID + col4 * 4);
        *(float4*)(&As[row * HID + col4 * 4]) = v;
    }
    __syncthreads();

    const int wave = tid >> 5;
    const int lane = tid & 31;
    const int half = lane >> 4;               // 0: lanes 0-15, 1: lanes 16-31
    const int r    = lane & 15;
    const int ncol = wave * 16 + r;           // output column this lane serves

    // Preload B fragments for all 32 K-steps: B[k][n] = W[n][k].
    // Lane layout (ISA 7.12.2, 32-bit 16x4): half 0 -> K={k0,k0+1},
    // half 1 -> K={k0+2,k0+3}; consecutive K pair = one 8-byte load.
    v2f breg[32];
    const float* wrow = W + (size_t)ncol * HID + 2 * half;
    #pragma unroll
    for (int s = 0; s < 32; ++s)
        breg[s] = *(const v2f*)(wrow + 4 * s);

    v8f acc = {};
    const float* arow = As + r * HID + 2 * half;
    #pragma unroll
    for (int s = 0; s < 32; ++s) {
        v2f a = *(const v2f*)(arow + 4 * s);   // ds_load_b64
        acc = __builtin_amdgcn_wmma_f32_16x16x4_f32(
                  false, a, false, breg[s], (short)0, acc, false, false);
    }

    // D layout: VGPR v -> M = v + 8*half (lanes 0-15 M=0..7, lanes 16-31 M=8..15),
    // N = lane & 15.
    #pragma unroll
    for (int v = 0; v < 8; ++v) {
        const int grow = m0 + 8 * half + v;
        if (grow < nrows)
            y[(size_t)grow * HID + ncol] = acc[v];
    }
}

// ---------------------------------------------------------------------------
// Phase 2: out[e] = sigmoid( z[src[e]] . y[dst[e]] )
// One wave32 per edge; each lane owns 4 contiguous floats (float4) of each
// 512-byte row -> fully coalesced; wave shuffle reduction. z and y (102 MB)
// both live in the 192 MB L2, so the random gathers are L2-resident.
// ---------------------------------------------------------------------------
__global__ __launch_bounds__(256)
void distmult_edge_dot(const float* __restrict__ z,
                       const float* __restrict__ y,
                       const int* __restrict__ src,
                       const int* __restrict__ dst,
                       float* __restrict__ out, int E)
{
    const int lane = threadIdx.x & 31;
    const int gid  = (blockIdx.x * blockDim.x + threadIdx.x) >> 5;  // edge id
    if (gid >= E) return;

    const int s = src[gid];
    const int d = dst[gid];
    const float4 a = *(const float4*)(z + (size_t)s * HID + lane * 4);
    const float4 b = *(const float4*)(y + (size_t)d * HID + lane * 4);
    float p = a.x * b.x + a.y * b.y + a.z * b.z + a.w * b.w;

    #pragma unroll
    for (int off = 16; off > 0; off >>= 1)
        p += __shfl_down(p, off, 32);

    if (lane == 0)
        out[gid] = 1.0f / (1.0f + __expf(-p));
}

// ---------------------------------------------------------------------------
// Fallback (only if ws_size can't hold y): naive fused bilinear per edge.
// ---------------------------------------------------------------------------
__global__ __launch_bounds__(128)
void distmult_fused_naive(const float* __restrict__ z,
                          const float* __restrict__ W,
                          const int* __restrict__ src,
                          const int* __restrict__ dst,
                          float* __restrict__ out, int E)
{
    const int t = blockIdx.x * blockDim.x + threadIdx.x;
    if (t >= E) return;
    const float* zs = z + (size_t)src[t] * HID;
    const float* zd = z + (size_t)dst[t] * HID;

    float4 dreg[32];
    #pragma unroll
    for (int j = 0; j < 32; ++j) dreg[j] = *(const float4*)(zd + 4 * j);

    float acc = 0.f;
    for (int i = 0; i < HID; ++i) {
        const float* wr = W + (size_t)i * HID;   // uniform -> scalar loads
        float h = 0.f;
        #pragma unroll
        for (int j = 0; j < 32; ++j) {
            const float4 w4 = *(const float4*)(wr + 4 * j);
            h += w4.x * dreg[j].x + w4.y * dreg[j].y
               + w4.z * dreg[j].z + w4.w * dreg[j].w;
        }
        acc += zs[i] * h;
    }
    out[t] = 1.0f / (1.0f + __expf(-acc));
}

extern "C" void kernel_launch(void* const* d_in, const int* in_sizes, int n_in,
                              void* d_out, int out_size, void* d_ws, size_t ws_size,
                              hipStream_t stream)
{
    const float* z  = (const float*)d_in[0];
    const int*   ei = (const int*)d_in[1];     // edge_index [2, E]
    const float* W  = (const float*)d_in[2];
    float*       out = (float*)d_out;

    const int N = in_sizes[0] / HID;
    const int E = in_sizes[1] / 2;
    const int* src = ei;
    const int* dst = ei + E;

    const size_t yBytes = (size_t)N * HID * sizeof(float);
    if (ws_size >= yBytes) {
        float* y = (float*)d_ws;
        const int rowTiles = (N + 15) / 16;
        distmult_gemm_zWT<<<rowTiles, 256, 0, stream>>>(z, W, y, N);

        const int edgesPerBlock = 8;           // 8 waves per 256-thread block
        const int blocks = (E + edgesPerBlock - 1) / edgesPerBlock;
        distmult_edge_dot<<<blocks, 256, 0, stream>>>(z, y, src, dst, out, E);
    } else {
        distmult_fused_naive<<<(E + 127) / 128, 128, 0, stream>>>(z, W, src, dst, out, E);
    }
}